// TriPlane_44418551775632
// MI455X (gfx1250) — compile-verified
//
#include <hip/hip_runtime.h>
#include <hip/hip_bf16.h>

#define RES   256
#define FEAT  64
#define PPB   16                 // points per block
#define TASKS (PPB * 3)          // (point, plane) tasks per block

// 8-byte vector load with only 4-byte alignment guarantee (tap pairs are
// float-aligned, not float2-aligned). gfx1250 supports unaligned global access.
typedef float f2v __attribute__((ext_vector_type(2), aligned(4)));

__global__ __launch_bounds__(256) void triplane_sample_kernel(
    const float* __restrict__ x,
    const float* __restrict__ pxy,
    const float* __restrict__ pxz,
    const float* __restrict__ pyz,
    float* __restrict__ out,
    int nPoints)
{
    __shared__ float  sCoord[PPB * 3];   // 48 floats of point coords
    __shared__ int2   sOff[TASKS];       // byte offsets of top/bottom tap pairs
    __shared__ float4 sW[TASKS];         // weights (top.l, top.r, bot.l, bot.r)

    const int tid   = threadIdx.x;
    const int pBase = blockIdx.x * PPB;

    // ---- Phase 0: async DMA this block's 16 point coords (48 floats) to LDS.
    if (tid < (PPB * 3) / 4) {           // 12 lanes x 16B
        long long gmax = (long long)nPoints * 3 - 4;
        long long gi   = (long long)pBase * 3 + tid * 4;
        if (gi > gmax) gi = gmax;        // tail clamp (still valid coords)
        unsigned long long gaddr = (unsigned long long)(const void*)(x + gi);
        unsigned ldsAddr = (unsigned)(size_t)(&sCoord[0]) + (unsigned)tid * 16u;
        asm volatile("global_load_async_to_lds_b128 %0, %1, off"
                     :: "v"(ldsAddr), "v"(gaddr) : "memory");
    }
    asm volatile("s_wait_asynccnt 0" ::: "memory");   // nop for non-issuing waves
    __syncthreads();

    // ---- Phase 1: 48 threads compute tap pairs, one (point,plane) each.
    if (tid < TASKS) {
        const int pt = tid / 3;
        const int pl = tid - pt * 3;                  // 0=xy, 1=xz, 2=yz
        const float cx = sCoord[pt * 3 + 0];
        const float cy = sCoord[pt * 3 + 1];
        const float cz = sCoord[pt * 3 + 2];
        const float u = (pl == 2) ? cy : cx;          // uv[:,0] -> ix (W dim)
        const float v = (pl == 0) ? cy : cz;          // uv[:,1] -> iy (H dim)

        const float ix  = (u + 1.0f) * 0.5f * (float)(RES - 1);
        const float iy  = (v + 1.0f) * 0.5f * (float)(RES - 1);
        const float fx0 = floorf(ix);
        const float fy0 = floorf(iy);
        // exact reference weight formulas (from UNCLAMPED floor coords)
        const float wxe = ix - fx0,  wxw = fx0 + 1.0f - ix;
        const float wys = iy - fy0,  wyn = fy0 + 1.0f - iy;

        const int x0 = (int)fminf(fmaxf(fx0, 0.0f), (float)(RES - 1));
        const int y0 = (int)fminf(fmaxf(fy0, 0.0f), (float)(RES - 1));
        const int y1 = (int)fminf(fmaxf(fy0 + 1.0f, 0.0f), (float)(RES - 1));

        // Pair trick: load floats [px, px+1]; if x0 was clamped to 255 then
        // ix==255.0 exactly so wxe==0, and index 255 carries weight wxw.
        const bool  edge = (x0 == RES - 1);
        const int   px   = edge ? (RES - 2) : x0;
        const float wl   = edge ? 0.0f : wxw;
        const float wr   = edge ? wxw  : wxe;

        sOff[tid] = make_int2((y0 * RES + px) * 4, (y1 * RES + px) * 4);
        sW[tid]   = make_float4(wl * wyn, wr * wyn, wl * wys, wr * wys);
    }
    __syncthreads();

    // ---- Phase 2: gather. Uniform SGPR plane base + 32-bit per-lane byte
    // offset (c*256KB + tap offset < 2^25) -> saddr-form global_load_b64.
    const int c  = tid & (FEAT - 1);
    const int pg = tid >> 6;                          // 0..3
    const unsigned coff = (unsigned)c << 18;          // c * 256*256*4 bytes
    const char* bXY = (const char*)pxy;
    const char* bXZ = (const char*)pxz;
    const char* bYZ = (const char*)pyz;

    #pragma unroll
    for (int s = 0; s < PPB / 4; ++s) {
        const int pt = pg * (PPB / 4) + s;
        const int t0 = pt * 3;
        const int2   oA = sOff[t0 + 0]; const float4 wA = sW[t0 + 0];
        const int2   oB = sOff[t0 + 1]; const float4 wB = sW[t0 + 1];
        const int2   oC = sOff[t0 + 2]; const float4 wC = sW[t0 + 2];

        const f2v tA = *(const f2v*)(bXY + (coff + (unsigned)oA.x));
        const f2v bA = *(const f2v*)(bXY + (coff + (unsigned)oA.y));
        const f2v tB = *(const f2v*)(bXZ + (coff + (unsigned)oB.x));
        const f2v bB = *(const f2v*)(bXZ + (coff + (unsigned)oB.y));
        const f2v tC = *(const f2v*)(bYZ + (coff + (unsigned)oC.x));
        const f2v bC = *(const f2v*)(bYZ + (coff + (unsigned)oC.y));

        float acc;
        acc  = tA.x * wA.x + tA.y * wA.y + bA.x * wA.z + bA.y * wA.w;
        acc += tB.x * wB.x + tB.y * wB.y + bB.x * wB.z + bB.y * wB.w;
        acc += tC.x * wC.x + tC.y * wC.y + bC.x * wC.z + bC.y * wC.w;

        const int p = pBase + pt;
        if (p < nPoints) {
            __builtin_nontemporal_store(acc * (1.0f / 3.0f),
                                        &out[(size_t)p * FEAT + c]);
        }
    }
}

extern "C" void kernel_launch(void* const* d_in, const int* in_sizes, int n_in,
                              void* d_out, int out_size, void* d_ws, size_t ws_size,
                              hipStream_t stream) {
    (void)n_in; (void)out_size; (void)d_ws; (void)ws_size;
    const float* x   = (const float*)d_in[0];   // [N,3]
    const float* pxy = (const float*)d_in[1];   // [1,64,256,256]
    const float* pxz = (const float*)d_in[2];
    const float* pyz = (const float*)d_in[3];
    // d_in[4] = pose (unused), d_in[5] = person_id (unused)
    float* out = (float*)d_out;                 // [N,64]

    const int nPoints = in_sizes[0] / 3;
    const int blocks  = (nPoints + PPB - 1) / PPB;
    triplane_sample_kernel<<<blocks, 256, 0, stream>>>(x, pxy, pxz, pyz, out, nPoints);
}